// BVH_37649683317091
// MI455X (gfx1250) — compile-verified
//
#include <hip/hip_runtime.h>
#include <hip/hip_bf16.h>
#include <stdint.h>

// Problem constants (match reference setup_inputs)
#define NBATCH   2
#define NTRI     4096
#define NPTS     16384
#define TILE     512                 // triangles per LDS tile
#define NTILES   (NTRI / TILE)       // 8
#define TILE_FLT (TILE * 9)          // 4608 floats per tile
#define TILE_B   (TILE_FLT * 4)      // 18432 bytes
#define THREADS  128                 // 4 wave32 per block
#define NWAVES   (THREADS / 32)      // 4
#define SLICE_FLT (TILE_FLT / NWAVES)            // 1152 floats per wave slice
#define SLICE_B   (SLICE_FLT * 4)                // 4608 bytes
#define CHUNKS_PER_THREAD ((TILE_B / 16) / THREADS)   // 9 (async fallback)

#if __has_builtin(__builtin_amdgcn_tensor_load_to_lds)
#define USE_TDM 1
#endif

typedef unsigned int v4u __attribute__((ext_vector_type(4)));
typedef int          v8i __attribute__((ext_vector_type(8)));
typedef int          v4i __attribute__((ext_vector_type(4)));

__device__ __forceinline__ float safe_div(float n, float d) {
    float ds = (__builtin_fabsf(d) > 1e-12f) ? d : 1.0f;
    return n / ds;
}
__device__ __forceinline__ float clamp01(float v) {
    return fminf(fmaxf(v, 0.0f), 1.0f);
}

#ifdef USE_TDM
// One Tensor-Data-Mover DMA per wave: copy a contiguous 1152-float (4608 B)
// slice from global to LDS. D# per CDNA5 ISA.
__device__ __forceinline__ void tdm_load_slice(uint32_t lds_off, uint64_t gaddr) {
    v4u g0;
    g0[0] = 1u;                                      // count=1, user descriptor
    g0[1] = lds_off;                                 // lds_addr (bytes)
    g0[2] = (uint32_t)(gaddr & 0xFFFFFFFFu);         // global_addr[31:0]
    g0[3] = ((uint32_t)(gaddr >> 32) & 0x01FFFFFFu)  // global_addr[56:32]
            | 0x80000000u;                           // type=2 ("image")
    v8i g1;
    g1[0] = (int)(2u << 16);                 // workgroup_mask=0, data_size=4B
    g1[1] = (int)((uint32_t)SLICE_FLT << 16);// tensor_dim0[15:0]=1152 (abar addr=0)
    g1[2] = (int)(1u << 16);                 // tensor_dim0[31:16]=0, tensor_dim1=1
    g1[3] = (int)((uint32_t)SLICE_FLT << 16);// tensor_dim1[31:16]=0, tile_dim0=1152
    g1[4] = 0;                               // tile_dim1=0, tile_dim2=0 (1D tile)
    g1[5] = (int)SLICE_FLT;                  // tensor_dim0_stride[31:0]
    g1[6] = 0;                               // stride hi / dim1_stride lo
    g1[7] = 0;                               // dim1_stride hi
    v4i z4 = (v4i)(0);
#if __clang_major__ >= 23
    v8i z8 = (v8i)(0);
    __builtin_amdgcn_tensor_load_to_lds(g0, g1, z4, z4, z8, 0);
#else
    __builtin_amdgcn_tensor_load_to_lds(g0, g1, z4, z4, 0);
#endif
}
#endif

// Copy one 18KB tile (512 tris * 9 f32) global -> LDS.
__device__ __forceinline__ void copy_tile(const float* __restrict__ gsrc,
                                          float* lds_dst, int tid) {
#ifdef USE_TDM
    const int wave = tid >> 5;                       // 0..3, wave-uniform
    uint32_t lds_off = (uint32_t)(uintptr_t)lds_dst + (uint32_t)wave * SLICE_B;
    uint64_t ga = (uint64_t)(uintptr_t)(gsrc + (size_t)wave * SLICE_FLT);
    tdm_load_slice(lds_off, ga);                     // one TDM op per wave
#else
    // Fallback: per-lane async loads (GLOBAL_LOAD_ASYNC_TO_LDS_B128, ASYNCcnt)
    const char* gbase = (const char*)gsrc;
    uint32_t lbase = (uint32_t)(uintptr_t)lds_dst;
#pragma unroll
    for (int k = 0; k < CHUNKS_PER_THREAD; ++k) {
        int chunk = tid + k * THREADS;
        uint64_t ga = (uint64_t)(uintptr_t)(gbase + chunk * 16);
        uint32_t la = lbase + chunk * 16;
        asm volatile("global_load_async_to_lds_b128 %0, %1, off"
                     :: "v"(la), "v"(ga) : "memory");
    }
#endif
}

__device__ __forceinline__ void wait_copy() {
#ifdef USE_TDM
    __builtin_amdgcn_s_wait_tensorcnt(0);            // s_wait_tensorcnt 0
#else
    asm volatile("s_wait_asynccnt 0x0" ::: "memory");
#endif
}

__global__ __launch_bounds__(THREADS)
void BVH_closest_point_kernel(const float* __restrict__ tris,   // [2,4096,3,3]
                              const float* __restrict__ pts,    // [2,16384,3]
                              float* __restrict__ dist_out,     // [2,16384]
                              float* __restrict__ cp_out,       // [2,16384,3]
                              int*   __restrict__ face_out) {   // [2,16384]
    __shared__ float sbuf[2][TILE_FLT];              // 36 KB double buffer

    const int tid = threadIdx.x;
    const int gid = blockIdx.x * THREADS + tid;      // 0..32767
    const int b   = gid >> 14;                       // / 16384 (uniform per block)
    const float* tb = tris + (size_t)b * NTRI * 9;

    const float px = pts[gid * 3 + 0];
    const float py = pts[gid * 3 + 1];
    const float pz = pts[gid * 3 + 2];

    // Prologue: DMA tile 0 into buffer 0.
    copy_tile(tb, sbuf[0], tid);
    wait_copy();
    __syncthreads();

    float best = INFINITY;
    int   bidx = 0;
    float bx = 0.f, by = 0.f, bz = 0.f;

    for (int t = 0; t < NTILES; ++t) {
        // Overlap: TDM DMAs the next tile while the VALUs chew on this one.
        if (t + 1 < NTILES)
            copy_tile(tb + (size_t)(t + 1) * TILE_FLT, sbuf[(t + 1) & 1], tid);

        const float* __restrict__ buf = sbuf[t & 1];
        for (int f = 0; f < TILE; ++f) {
            const float* tr = buf + f * 9;           // uniform addr -> LDS broadcast
            const float ax = tr[0], ay = tr[1], az = tr[2];
            const float bvx = tr[3], bvy = tr[4], bvz = tr[5];
            const float cx = tr[6], cy = tr[7], cz = tr[8];

            const float abx = bvx - ax, aby = bvy - ay, abz = bvz - az;
            const float acx = cx - ax,  acy = cy - ay,  acz = cz - az;
            const float apx = px - ax,  apy = py - ay,  apz = pz - az;
            const float dd1 = abx*apx + aby*apy + abz*apz;
            const float dd2 = acx*apx + acy*apy + acz*apz;
            const float bpx = px - bvx, bpy = py - bvy, bpz = pz - bvz;
            const float dd3 = abx*bpx + aby*bpy + abz*bpz;
            const float dd4 = acx*bpx + acy*bpy + acz*bpz;
            const float cpx = px - cx,  cpy = py - cy,  cpz = pz - cz;
            const float dd5 = abx*cpx + aby*cpy + abz*cpz;
            const float dd6 = acx*cpx + acy*cpy + acz*cpz;

            const float vc = dd1*dd4 - dd3*dd2;
            const float vb = dd5*dd2 - dd1*dd6;
            const float va = dd3*dd6 - dd5*dd4;

            const float v_ab = clamp01(safe_div(dd1, dd1 - dd3));
            const float w_ac = clamp01(safe_div(dd2, dd2 - dd6));
            const float w_bc = clamp01(safe_div(dd4 - dd3, (dd4 - dd3) + (dd5 - dd6)));

            const float denom = va + vb + vc;
            const float vv = safe_div(vb, denom);
            const float ww = safe_div(vc, denom);

            float rx = ax + abx * vv + acx * ww;
            float ry = ay + aby * vv + acy * ww;
            float rz = az + abz * vv + acz * ww;

            // Same override order as the reference's jnp.where chain.
            const bool m_bc = (va <= 0.f) && (dd4 - dd3 >= 0.f) && (dd5 - dd6 >= 0.f);
            const bool m_ac = (vb <= 0.f) && (dd2 >= 0.f) && (dd6 <= 0.f);
            const bool m_ab = (vc <= 0.f) && (dd1 >= 0.f) && (dd3 <= 0.f);
            const bool m_c  = (dd6 >= 0.f) && (dd5 <= dd6);
            const bool m_b  = (dd3 >= 0.f) && (dd4 <= dd3);
            const bool m_a  = (dd1 <= 0.f) && (dd2 <= 0.f);

            if (m_bc) { rx = bvx + w_bc*(cx-bvx); ry = bvy + w_bc*(cy-bvy); rz = bvz + w_bc*(cz-bvz); }
            if (m_ac) { rx = ax + w_ac*acx; ry = ay + w_ac*acy; rz = az + w_ac*acz; }
            if (m_ab) { rx = ax + v_ab*abx; ry = ay + v_ab*aby; rz = az + v_ab*abz; }
            if (m_c)  { rx = cx; ry = cy; rz = cz; }
            if (m_b)  { rx = bvx; ry = bvy; rz = bvz; }
            if (m_a)  { rx = ax; ry = ay; rz = az; }

            const float dx = px - rx, dy = py - ry, dz = pz - rz;
            const float d2v = dx*dx + dy*dy + dz*dz;
            if (d2v < best) {            // strict: keep first argmin like jnp.argmin
                best = d2v; bidx = t * TILE + f;
                bx = rx; by = ry; bz = rz;
            }
        }

        if (t + 1 < NTILES) {
            wait_copy();                 // this wave's slice of tile t+1 landed
            __syncthreads();             // all slices landed; tile-t reads retired
        }
    }

    dist_out[gid] = best;
    cp_out[gid * 3 + 0] = bx;
    cp_out[gid * 3 + 1] = by;
    cp_out[gid * 3 + 2] = bz;
    face_out[gid] = bidx;
}

extern "C" void kernel_launch(void* const* d_in, const int* in_sizes, int n_in,
                              void* d_out, int out_size, void* d_ws, size_t ws_size,
                              hipStream_t stream) {
    const float* tris = (const float*)d_in[0];   // 2*4096*3*3 f32
    const float* pts  = (const float*)d_in[1];   // 2*16384*3  f32

    float* out   = (float*)d_out;
    float* dist  = out;                                  // 32768 f32
    float* cp    = out + NBATCH * NPTS;                  // 98304 f32
    int*   faces = (int*)(out + NBATCH * NPTS + NBATCH * NPTS * 3); // 32768 i32

    const int total_pts = NBATCH * NPTS;                 // 32768
    const int blocks = total_pts / THREADS;              // 256
    BVH_closest_point_kernel<<<blocks, THREADS, 0, stream>>>(tris, pts, dist, cp, faces);
}